// Assigner_72353019068756
// MI455X (gfx1250) — compile-verified
//
#include <hip/hip_runtime.h>
#include <stdint.h>

// 3D IoU-threshold assigner (MaxIoUAssigner-style), N anchors vs M=64 gts.
// Outputs (concatenated in d_out, float): labels[N], assigned_bboxes[N*6].
//
// MI455X / gfx1250 strategy:
//  - VALU-issue-bound (~24 fp32 ops per anchor-gt pair, ~370M lane-ops vs
//    ~14MB HBM traffic). Inner loop is fully branchless (first-hit index via
//    v_cndmask, any/all via mask ops); winning gt record is fetched once
//    after the loop.
//  - FOUR anchors per thread sharing each gt record: four independent
//    dependency chains feed VOPD dual-issue (v_dual_max/min/mul/add/cndmask
//    are pairable), and LDS broadcast traffic drops to 0.5 ds_load_b128 per
//    anchor per gt.
//  - Division-free thresholds: iou>=0.5 <=> 2*inter>=denom,
//    iou<0.3 <=> inter<0.3*denom (denom>0 due to eps). gt volume precomputed.
//  - gt boxes+labels staged via CDNA5 async global->LDS (ASYNCcnt path),
//    repacked to 32B records. Unconditional (clamped) prefetches -- no
//    EXEC-mask branches in the prologue.

#define MGT   64
#define BLOCK 256
#define APT   4      // anchors per thread

__device__ __forceinline__ uint32_t lds_off_of(const void* p) {
  // Flat LDS aperture keeps the workgroup-relative byte offset in addr[31:0].
  return (uint32_t)(uintptr_t)p;
}

__global__ __launch_bounds__(BLOCK) void assign3d_kernel(
    const float* __restrict__ bboxes,     // [N,6]
    const float* __restrict__ gt_bboxes,  // [64,6]
    const int*   __restrict__ gt_labels,  // [64]
    float* __restrict__ out_labels,       // [N]
    float* __restrict__ out_bboxes,       // [N,6]
    int N)
{
  __shared__ __align__(16) float s_box[MGT * 6];   // packed staging (1536 B)
  __shared__ __align__(16) int   s_labi[MGT];      // label staging (256 B)
  __shared__ __align__(16) float s_rec[MGT * 8];   // [z1,y1,x1,z2,y2,x2,vol2,label]

  const int t    = threadIdx.x;
  const int base = blockIdx.x * (BLOCK * APT) + t;

  // Branchless prefetch of this thread's anchor rows (clamped; prefetch is
  // side-effect free so no EXEC guarding needed).
#pragma unroll
  for (int a = 0; a < APT; ++a) {
    int pi = min(base + a * BLOCK, N - 1);
    __builtin_prefetch(bboxes + 6 * (size_t)pi, 0, 3);
  }

#if defined(__gfx1250__)
  // CDNA5 async global->LDS: each enabled lane moves 16B (ASYNCcnt-tracked).
  if (t < (MGT * 24) / 16) {                          // 96 chunks of boxes
    uint32_t l = lds_off_of(s_box) + (uint32_t)t * 16u;
    uint64_t g = (uint64_t)(uintptr_t)gt_bboxes + (uint64_t)t * 16u;
    asm volatile("global_load_async_to_lds_b128 %0, %1, off"
                 :: "v"(l), "v"(g) : "memory");
  } else if (t < (MGT * 24) / 16 + (MGT * 4) / 16) {  // 16 chunks of labels
    uint32_t c = (uint32_t)t - (MGT * 24) / 16;
    uint32_t l = lds_off_of(s_labi) + c * 16u;
    uint64_t g = (uint64_t)(uintptr_t)gt_labels + (uint64_t)c * 16u;
    asm volatile("global_load_async_to_lds_b128 %0, %1, off"
                 :: "v"(l), "v"(g) : "memory");
  }
#if __has_builtin(__builtin_amdgcn_s_wait_asynccnt)
  __builtin_amdgcn_s_wait_asynccnt(0);
#else
  asm volatile("s_wait_asynccnt 0x0" ::: "memory");
#endif
#else
  // Non-gfx1250 fallback: plain vector copies.
  if (t < 96)       ((float4*)s_box)[t]     = ((const float4*)gt_bboxes)[t];
  else if (t < 112) ((int4*)s_labi)[t - 96] = ((const int4*)gt_labels)[t - 96];
#endif
  __syncthreads();

  // Build padded 32B gt records incl. precomputed volume and float label.
  if (t < MGT) {
    const float* g = s_box + 6 * t;
    float gz1 = g[0], gy1 = g[1], gx1 = g[2];
    float gz2 = g[3], gy2 = g[4], gx2 = g[5];
    float vol2 = (gz2 - gz1) * (gy2 - gy1) * (gx2 - gx1);
    float* r = s_rec + 8 * t;
    r[0] = gz1; r[1] = gy1; r[2] = gx1; r[3] = gz2; r[4] = gy2; r[5] = gx2;
    r[6] = vol2; r[7] = (float)s_labi[t];
  }
  __syncthreads();

  // ---- per-thread anchor state: APT independent streams for dual-issue ----
  int   idx[APT];
  float az1[APT], ay1[APT], ax1[APT], az2[APT], ay2[APT], ax2[APT], voleps[APT];
  bool  active[APT], pos[APT], allneg[APT];
  int   pm[APT];

#pragma unroll
  for (int a = 0; a < APT; ++a) {
    idx[a]    = base + a * BLOCK;
    active[a] = idx[a] < N;
    int li    = active[a] ? idx[a] : 0;          // clamp; store is guarded
    const float2* bp = (const float2*)(bboxes + 6 * (size_t)li);
    float2 p0 = bp[0], p1 = bp[1], p2 = bp[2];   // 3x b64, coalesced
    az1[a] = p0.x; ay1[a] = p0.y; ax1[a] = p1.x;
    az2[a] = p1.y; ay2[a] = p2.x; ax2[a] = p2.y;
    voleps[a] = (az2[a] - az1[a]) * (ay2[a] - ay1[a]) * (ax2[a] - ax1[a]) + 1e-7f;
    pos[a] = false; allneg[a] = true; pm[a] = 0;
  }

#pragma unroll 4
  for (int m = 0; m < MGT; ++m) {
    float4 r0 = *(const float4*)(s_rec + 8 * m);      // z1,y1,x1,z2 (broadcast)
    float4 r1 = *(const float4*)(s_rec + 8 * m + 4);  // y2,x2,vol2,label
#pragma unroll
    for (int a = 0; a < APT; ++a) {
      float dz = fminf(az2[a], r0.w) - fmaxf(az1[a], r0.x);
      float dy = fminf(ay2[a], r1.x) - fmaxf(ay1[a], r0.y);
      float dx = fminf(ax2[a], r1.y) - fmaxf(ax1[a], r0.z);
      dz = fmaxf(dz, 0.f); dy = fmaxf(dy, 0.f); dx = fmaxf(dx, 0.f);
      float inter = dz * dy * dx;
      float denom = voleps[a] + r1.z - inter;          // vol1+vol2-inter+eps > 0
      bool  hit   = (inter + inter) >= denom;          // iou >= 0.5
      if (hit && !pos[a]) pm[a] = m;                   // first True -> v_cndmask
      pos[a]    = pos[a] || hit;
      allneg[a] = allneg[a] && (inter < 0.3f * denom); // iou < 0.3
    }
  }

#pragma unroll
  for (int a = 0; a < APT; ++a) {
    if (!active[a]) continue;
    // Fetch the winning gt record once (divergent ds_load_b128 pair).
    float4 h0 = *(const float4*)(s_rec + 8 * pm[a]);
    float4 h1 = *(const float4*)(s_rec + 8 * pm[a] + 4);

    float lab, b0, b1, b2, b3, b4, b5;
    if (pos[a]) {
      lab = h1.w;
      b0 = h0.x; b1 = h0.y; b2 = h0.z; b3 = h0.w; b4 = h1.x; b5 = h1.y;
    } else {
      lab = allneg[a] ? 0.f : -1.f;
      b0 = b1 = b2 = b3 = b4 = b5 = -1.f;
    }

    out_labels[idx[a]] = lab;
    float2* op = (float2*)(out_bboxes + 6 * (size_t)idx[a]);
    op[0] = make_float2(b0, b1);
    op[1] = make_float2(b2, b3);
    op[2] = make_float2(b4, b5);
  }
}

extern "C" void kernel_launch(void* const* d_in, const int* in_sizes, int n_in,
                              void* d_out, int out_size, void* d_ws, size_t ws_size,
                              hipStream_t stream) {
  const float* bboxes    = (const float*)d_in[0];   // [N,6] f32
  const float* gt_bboxes = (const float*)d_in[1];   // [64,6] f32
  const int*   gt_labels = (const int*)d_in[2];     // [64] int
  const int N = in_sizes[0] / 6;

  float* out_labels = (float*)d_out;      // [N]
  float* out_bboxes = out_labels + N;     // [N,6]

  const int per_block = BLOCK * APT;
  dim3 grid((N + per_block - 1) / per_block), block(BLOCK);
  assign3d_kernel<<<grid, block, 0, stream>>>(bboxes, gt_bboxes, gt_labels,
                                              out_labels, out_bboxes, N);
}